// fusionLossFunc_improved_31782757990606
// MI455X (gfx1250) — compile-verified
//
#include <hip/hip_runtime.h>
#include <hip/hip_bf16.h>

// Fused landmark loss for MI455X (gfx1250, wave32).
// Algebraic restructuring: BCE(x,z) = softplus(x) - x*z, and heat/offsets are
// nonzero only within an 83x83 bounding box per landmark. So we stream only
// the 19 logit channels (77.8 MB) and touch pred channels only in the discs
// (~1 MB) -> ~2.9x less traffic than the naive 233 MB stream.
// Block reductions use V_WMMA_F32_16X16X4_F32 with an all-ones B matrix
// (tensor-core reduction; depends only on documented A and C/D layouts).
// Streaming kernel: statically 4 iterations -> fully unrolled, all four
// global_load_b128 (non-temporal) issued before any compute for deep MLP,
// with 4 independent accumulators to break the FP add dependency chain.

typedef float v2f __attribute__((ext_vector_type(2)));
typedef float v4f __attribute__((ext_vector_type(4)));
typedef float v8f __attribute__((ext_vector_type(8)));

#define HDIM 800
#define WDIM 640
#define HW   (HDIM * WDIM)      // 512000
#define LNUM 19
#define BNUM 2
#define CHN  (BNUM * LNUM)      // 38 channels
#define CSTRIDE (3 * LNUM)      // 57 channels per batch in featureMaps
#define BPC  125                // blocks per channel for the streaming kernel
#define TPB  256                // 8 waves of 32
#define STR  (BPC * TPB)        // 32000 float4s stride; HW/4 == 4*STR exactly
#define R1SQF 1681.0f           // 41^2
#define R2F   41.0f

// ---- wave32 reduction via WMMA f32 16x16x4 ----
// A per-lane = (v, 0): A[m,0]=v_m (lanes 0..15), A[m,2]=v_{m+16} (lanes 16..31),
// A[m,1]=A[m,3]=0.  B = ones.  D[m,n] = v_m + v_{m+16} for every n.
// C/D layout: VGPR r, lanes 0..15 -> D[r,n]; lanes 16..31 -> D[r+8,n].
// So sum(d[0..7]) gives rowsums 0..7 in lanes 0..15 and rowsums 8..15 in
// lanes 16..31; one xor-16 shuffle completes the full 32-lane sum.
__device__ __forceinline__ float wave_reduce_wmma(float v) {
    v2f a; a[0] = v;    a[1] = 0.0f;
    v2f b; b[0] = 1.0f; b[1] = 1.0f;
    v8f c = {};
    v8f d = __builtin_amdgcn_wmma_f32_16x16x4_f32(
        /*neg_a=*/false, a, /*neg_b=*/false, b,
        /*c_mod=*/(short)0, c, /*reuse_a=*/false, /*reuse_b=*/false);
    float s = ((d[0] + d[1]) + (d[2] + d[3])) + ((d[4] + d[5]) + (d[6] + d[7]));
    s += __shfl_xor(s, 16, 32);
    return s;                       // all lanes hold the wave sum
}

// Full-block (256 threads = 8 waves) sum; result valid on threadIdx.x == 0.
// Must be called with all threads of the block converged (EXEC all ones).
__device__ __forceinline__ float block_reduce(float v, float* lds) {
    float ws = wave_reduce_wmma(v);
    const unsigned tid = threadIdx.x;
    if ((tid & 31u) == 0u) lds[tid >> 5] = ws;
    __syncthreads();
    float r = 0.0f;
    if (tid == 0) {
#pragma unroll
        for (int w = 0; w < TPB / 32; ++w) r += lds[w];
    }
    __syncthreads();
    return r;
}

__device__ __forceinline__ float softplus_fast(float x) {
    // log1p(exp(-|x|)) + max(x,0); for |x| in the N(0,1) range exp(-|x|) >= e^-6,
    // so log(1 + e) via hardware v_log/v_exp loses no meaningful precision.
    return fmaxf(x, 0.0f) + __logf(1.0f + __expf(-fabsf(x)));
}

__device__ __forceinline__ float softplus4(v4f x) {
    return (softplus_fast(x[0]) + softplus_fast(x[1])) +
           (softplus_fast(x[2]) + softplus_fast(x[3]));
}

// ---- Kernel A: stream all logit channels, per-block softplus partial sums ----
__global__ void __launch_bounds__(TPB)
stream_softplus_kernel(const float* __restrict__ fm, float* __restrict__ partials) {
    __shared__ float lds[TPB / 32];
    const int blk = blockIdx.x;
    const int ch  = blk / BPC;          // 0..37
    const int sub = blk - ch * BPC;     // 0..124
    const int b   = ch / LNUM;
    const int l   = ch - b * LNUM;
    const v4f* __restrict__ src =
        (const v4f*)(fm + (size_t)(b * CSTRIDE + l) * HW);

    const int base = sub * TPB + (int)threadIdx.x;   // 0..31999
    // Issue all four 128-bit non-temporal loads before any compute.
    v4f x0 = __builtin_nontemporal_load(src + base);
    v4f x1 = __builtin_nontemporal_load(src + base + STR);
    v4f x2 = __builtin_nontemporal_load(src + base + 2 * STR);
    v4f x3 = __builtin_nontemporal_load(src + base + 3 * STR);

    const float a0 = softplus4(x0);
    const float a1 = softplus4(x1);
    const float a2 = softplus4(x2);
    const float a3 = softplus4(x3);
    const float acc = (a0 + a1) + (a2 + a3);

    float s = block_reduce(acc, lds);
    if (threadIdx.x == 0) partials[blk] = s;
}

// ---- Kernel B: per-landmark disc sums (logit correction, L1 terms, mask count) ----
__global__ void __launch_bounds__(TPB)
disc_kernel(const float* __restrict__ fm, const float* __restrict__ lm,
            float* __restrict__ disc) {
    __shared__ float lds[TPB / 32];
    const int ch = blockIdx.x;          // 0..37
    const int b  = ch / LNUM;
    const int l  = ch - b * LNUM;

    // jnp.round == round-half-even == rintf under default RTNE
    const float X = rintf(lm[(b * LNUM + l) * 2 + 0] * (float)(HDIM - 1));
    const float Y = rintf(lm[(b * LNUM + l) * 2 + 1] * (float)(WDIM - 1));
    const int xi = (int)X, yi = (int)Y;
    const int i0 = max(xi - 41, 0), i1 = min(xi + 41, HDIM - 1);
    const int j0 = max(yi - 41, 0), j1 = min(yi + 41, WDIM - 1);
    const int nJ  = j1 - j0 + 1;
    const int tot = (i1 - i0 + 1) * nJ;

    const float* __restrict__ lg = fm + (size_t)(b * CSTRIDE + l) * HW;
    const float* __restrict__ px = fm + (size_t)(b * CSTRIDE + LNUM + l) * HW;
    const float* __restrict__ py = fm + (size_t)(b * CSTRIDE + 2 * LNUM + l) * HW;

    float sx = 0.0f, sl1x = 0.0f, sl1y = 0.0f, cnt = 0.0f;
    for (int t = threadIdx.x; t < tot; t += TPB) {
        const int ti = t / nJ;
        const int i  = i0 + ti;
        const int j  = j0 + (t - ti * nJ);
        const float dx = X - (float)i;   // integer-valued f32: exact arithmetic
        const float dy = Y - (float)j;
        if (dx * dx + dy * dy <= R1SQF) {
            const int o = i * WDIM + j;
            sx   += lg[o];
            sl1x += fabsf(px[o] - dx / R2F);
            sl1y += fabsf(py[o] - dy / R2F);
            cnt  += 1.0f;
        }
    }
    float r;
    r = block_reduce(sx,   lds); if (threadIdx.x == 0) disc[ch * 4 + 0] = r;
    r = block_reduce(sl1x, lds); if (threadIdx.x == 0) disc[ch * 4 + 1] = r;
    r = block_reduce(sl1y, lds); if (threadIdx.x == 0) disc[ch * 4 + 2] = r;
    r = block_reduce(cnt,  lds); if (threadIdx.x == 0) disc[ch * 4 + 3] = r;
}

// ---- Kernel C: combine per-channel sums into the scalar loss ----
__global__ void __launch_bounds__(TPB)
finalize_kernel(const float* __restrict__ partials, const float* __restrict__ disc,
                float* __restrict__ out) {
    __shared__ float lds[TPB / 32];
    const int t = threadIdx.x;
    float acc = 0.0f;
    if (t < CHN) {
        float sp = 0.0f;
        const float* p = partials + t * BPC;
        for (int k = 0; k < BPC; ++k) sp += p[k];
        const float sx  = disc[t * 4 + 0];
        const float l1x = disc[t * 4 + 1];
        const float l1y = disc[t * 4 + 2];
        const float cnt = disc[t * 4 + 3];
        // bce[b,l] = 2/HW * (sum softplus(x) - sum_disc x) ; then + l1x/m + l1y/m
        acc = (sp - sx) * (2.0f / (float)HW) + l1x / cnt + l1y / cnt;
    }
    float s = block_reduce(acc, lds);
    if (t == 0) out[0] = s * (1.0f / (float)CHN);
}

extern "C" void kernel_launch(void* const* d_in, const int* in_sizes, int n_in,
                              void* d_out, int out_size, void* d_ws, size_t ws_size,
                              hipStream_t stream) {
    (void)in_sizes; (void)n_in; (void)out_size; (void)ws_size;
    const float* fm = (const float*)d_in[0];   // featureMaps (2,57,800,640) f32
    const float* lm = (const float*)d_in[1];   // landmarks   (2,19,2) f32
    float* out = (float*)d_out;

    float* partials = (float*)d_ws;            // CHN*BPC floats (all overwritten)
    float* disc     = partials + CHN * BPC;    // CHN*4 floats   (all overwritten)

    stream_softplus_kernel<<<CHN * BPC, TPB, 0, stream>>>(fm, partials);
    disc_kernel<<<CHN, TPB, 0, stream>>>(fm, lm, disc);
    finalize_kernel<<<1, TPB, 0, stream>>>(partials, disc, out);
}